// TopK_31877247271346
// MI455X (gfx1250) — compile-verified
//
#include <hip/hip_runtime.h>

#ifndef __has_builtin
#define __has_builtin(x) 0
#endif

#define GLOBAL_AS __attribute__((address_space(1)))
#define LDS_AS    __attribute__((address_space(3)))

typedef int v4i_t __attribute__((ext_vector_type(4)));
typedef GLOBAL_AS v4i_t* gv4i_p;   // global (AS1) int4*
typedef LDS_AS    v4i_t* lv4i_p;   // LDS    (AS3) int4*

// ---------------------------------------------------------------------------
// CDNA5 async data-mover helpers (ASYNCcnt-tracked LDS<->global DMA).
// Builtin path when available; inline-asm fallback with gfx1250 mnemonics.
// ---------------------------------------------------------------------------
__device__ __forceinline__ void async_copy_g2l_b128(const float* g, float* l) {
#if __has_builtin(__builtin_amdgcn_global_load_async_to_lds_b128)
  __builtin_amdgcn_global_load_async_to_lds_b128(
      (gv4i_p)(void*)(const_cast<float*>(g)), (lv4i_p)(void*)(l), 0, 0);
#else
  unsigned loff = (unsigned)(unsigned long long)(LDS_AS void*)(l);
  asm volatile("global_load_async_to_lds_b128 %0, %1, off"
               :: "v"(loff), "v"((unsigned long long)(const void*)g)
               : "memory");
#endif
}

__device__ __forceinline__ void async_copy_l2g_b128(float* g, const float* l) {
#if __has_builtin(__builtin_amdgcn_global_store_async_from_lds_b128)
  __builtin_amdgcn_global_store_async_from_lds_b128(
      (gv4i_p)(void*)(g), (lv4i_p)(void*)(const_cast<float*>(l)), 0, 0);
#else
  unsigned loff = (unsigned)(unsigned long long)(LDS_AS void*)(l);
  asm volatile("global_store_async_from_lds_b128 %0, %1, off"
               :: "v"((unsigned long long)(void*)g), "v"(loff)
               : "memory");
#endif
}

__device__ __forceinline__ void wait_async0() {
#if __has_builtin(__builtin_amdgcn_s_wait_asynccnt)
  __builtin_amdgcn_s_wait_asynccnt(0);
#else
  asm volatile("s_wait_asynccnt 0" ::: "memory");
#endif
}

// Order-preserving float -> uint key (larger float => larger uint).
__device__ __forceinline__ unsigned fkey(float v) {
  unsigned u = __float_as_uint(v);
  return u ^ ((u >> 31) ? 0xFFFFFFFFu : 0x80000000u);
}

// ---------------------------------------------------------------------------
// One workgroup per row. Row (64 KB) lives in LDS; 4-round radix select for
// the K-th largest; in-place keep/zero; async stream-out of the full row.
// ---------------------------------------------------------------------------
constexpr int D = 16384;
constexpr int K = 64;
constexpr int THREADS = 256;

__global__ __launch_bounds__(THREADS)
void topk_scatter_kernel(const float* __restrict__ x, float* __restrict__ out) {
  __shared__ float    rowv[D];      // 64 KB row buffer (of 320 KB WGP LDS)
  __shared__ unsigned hist[256];    // radix histogram
  __shared__ unsigned ctrl[4];      // [0]=digit, [1]=kremain, [2]=tie counter

  const int tid = threadIdx.x;
  const float* gx   = x   + (size_t)blockIdx.x * D;
  float*       gout = out + (size_t)blockIdx.x * D;

  // --- 1) DMA the row into LDS: 4096 x 16B chunks, 16 issues per thread ---
  #pragma unroll
  for (int i = 0; i < D / 4 / THREADS; ++i) {
    const int c = tid + i * THREADS;          // 16-byte chunk id
    async_copy_g2l_b128(gx + c * 4, rowv + c * 4);
  }
  wait_async0();
  __syncthreads();

  // --- 2) radix select: exact K-th largest key, MSB-first, 4 x 8-bit -------
  unsigned prefix = 0, pmask = 0, kremain = K;
  #pragma unroll
  for (int round = 0; round < 4; ++round) {
    const int shift = 24 - 8 * round;
    if (tid < 256) hist[tid] = 0;
    __syncthreads();

    for (int i = tid; i < D; i += THREADS) {
      const unsigned u = fkey(rowv[i]);
      if ((u & pmask) == prefix)
        atomicAdd(&hist[(u >> shift) & 0xFFu], 1u);
    }
    __syncthreads();

    if (tid == 0) {
      unsigned kr = kremain, cum = 0, digit = 0;
      for (int b = 255; b >= 0; --b) {
        const unsigned h = hist[b];
        if (cum + h >= kr) { digit = (unsigned)b; kr -= cum; break; }
        cum += h;
      }
      ctrl[0] = digit;
      ctrl[1] = kr;
    }
    __syncthreads();
    prefix |= ctrl[0] << shift;
    pmask  |= 0xFFu << shift;
    kremain = ctrl[1];
    __syncthreads();
  }

  // --- 3) in-place keep/zero in LDS (exact-K via tie quota) ----------------
  if (tid == 0) ctrl[2] = 0;
  __syncthreads();
  const unsigned T = prefix;        // key of the K-th largest element
  const unsigned quota = kremain;   // how many T-valued elements to keep
  for (int i = tid; i < D; i += THREADS) {
    const unsigned u = fkey(rowv[i]);
    bool keep = (u > T);
    if (u == T) {
      const unsigned slot = atomicAdd(&ctrl[2], 1u);
      keep = (slot < quota);
    }
    if (!keep) rowv[i] = 0.0f;
  }
  __syncthreads();

  // --- 4) DMA the finished row back out (single coalesced write pass) -----
  #pragma unroll
  for (int i = 0; i < D / 4 / THREADS; ++i) {
    const int c = tid + i * THREADS;
    async_copy_l2g_b128(gout + c * 4, rowv + c * 4);
  }
  wait_async0();
}

// ---------------------------------------------------------------------------
extern "C" void kernel_launch(void* const* d_in, const int* in_sizes, int n_in,
                              void* d_out, int out_size, void* d_ws, size_t ws_size,
                              hipStream_t stream) {
  (void)n_in; (void)out_size; (void)d_ws; (void)ws_size;
  const float* x  = (const float*)d_in[0];
  float* out      = (float*)d_out;
  const int rows  = in_sizes[0] / D;   // 4096
  topk_scatter_kernel<<<dim3(rows), dim3(THREADS), 0, stream>>>(x, out);
}